// GroupConvoution_23055384445550
// MI455X (gfx1250) — compile-verified
//
#include <hip/hip_runtime.h>

typedef __attribute__((ext_vector_type(2))) float v2f;
typedef __attribute__((ext_vector_type(4))) float v4f;
typedef __attribute__((ext_vector_type(8))) float v8f;
typedef __attribute__((ext_vector_type(4))) int   v4i;

#define LRELU_ALPHA 0.3f
#define BN_EPS 1e-3f

// Geometry (fixed by the reference problem)
#define BB 16
#define HH 64
#define WW 64
#define CC 1024
#define GG 32
#define CG 32
#define TH 8     // tile rows per block
#define TW 16    // tile cols per block
#define HR (TH + 2)
#define WR (TW + 2)

typedef __attribute__((address_space(1))) v4i* glb_v4i_ptr;
typedef __attribute__((address_space(3))) v4i* lds_v4i_ptr;

#if __has_builtin(__builtin_amdgcn_global_load_async_to_lds_b128) && \
    __has_builtin(__builtin_amdgcn_s_wait_asynccnt)
#define USE_ASYNC_LD 1
#else
#define USE_ASYNC_LD 0
#endif

#if __has_builtin(__builtin_amdgcn_global_store_async_from_lds_b128)
#define USE_ASYNC_ST 1
#else
#define USE_ASYNC_ST 0
#endif

#if __has_builtin(__builtin_amdgcn_rsqf)
#define RSQ(x) __builtin_amdgcn_rsqf(x)
#else
#define RSQ(x) rsqrtf(x)
#endif

__global__ __launch_bounds__(256)
void fused_dw_pw_bn_lrelu(const float* __restrict__ x,
                          const float* __restrict__ dwk,     // [3][3][1][C]
                          const float* __restrict__ pwk,     // [G][Cg][Cg]
                          const float* __restrict__ gamma_,  // [C]
                          const float* __restrict__ beta_,   // [C]
                          const float* __restrict__ mmean,   // [C]
                          const float* __restrict__ mvar,    // [C]
                          float* __restrict__ out)
{
    __shared__ float s_in[HR * WR * CG];       // halo input tile [r][c][ch]   23040 B
    __shared__ float s_dw[TH * TW * CG];       // dw result, then output tile  16384 B
    __shared__ float s_pw[(CG / 2) * CG * 2];  // K-pair-packed pw weights      4096 B

    const int tid    = threadIdx.x;
    const int tileId = blockIdx.x;          // 0..31  (8 row-tiles x 4 col-tiles)
    const int b      = blockIdx.y;          // batch
    const int g      = blockIdx.z;          // channel group
    const int th0    = (tileId / (WW / TW)) * TH;
    const int tw0    = (tileId % (WW / TW)) * TW;
    const int chBase = g * CG;

    // ---------------- stage halo input tile -------------------------------------
    // Interior lanes: async global->LDS b128 (ASYNCcnt path, no VGPR staging).
    // Zero-pad lanes: direct LDS store of zeros.
    const float* xg = x + (size_t)b * HH * WW * CC + chBase;
    for (int idx = tid; idx < HR * WR * (CG / 4); idx += 256) {
        int px = idx >> 3;              // 8 float4-quads per halo pixel
        int q  = idx & 7;
        int hr = th0 + (px / WR) - 1;
        int wc = tw0 + (px % WR) - 1;
        float* lp = &s_in[px * CG + q * 4];
        if (hr >= 0 && hr < HH && wc >= 0 && wc < WW) {
            const float* gp = xg + ((size_t)hr * WW + wc) * CC + q * 4;
#if USE_ASYNC_LD
            __builtin_amdgcn_global_load_async_to_lds_b128(
                (glb_v4i_ptr)gp, (lds_v4i_ptr)lp, 0, 0);
#else
            *(v4f*)lp = *(const v4f*)gp;
#endif
        } else {
            v4f z = {0.f, 0.f, 0.f, 0.f};
            *(v4f*)lp = z;
        }
    }

    // ---------------- stage pw weights, K-pair packed ----------------------------
    // s_pw[((c>>1)*CG + d)*2 + (c&1)] = pw[g][c][d]  ->  B fragment = one b64 load
    for (int idx = tid; idx < CG * CG; idx += 256) {
        int c = idx >> 5;
        int d = idx & 31;
        s_pw[((c >> 1) * CG + d) * 2 + (c & 1)] = pwk[g * CG * CG + idx];
    }

#if USE_ASYNC_LD
    __builtin_amdgcn_s_wait_asynccnt(0);
#endif
    __syncthreads();

    // ---------------- depthwise 3x3 ('SAME'), sliding window per row -------------
    {
        const int ch = tid & (CG - 1);
        const int r0 = tid >> 5;            // output row 0..7, one per thread slice
        float k[9];
#pragma unroll
        for (int i = 0; i < 9; ++i) k[i] = dwk[i * CC + chBase + ch];

        float accs[TW];
#pragma unroll
        for (int pc = 0; pc < TW; ++pc) accs[pc] = 0.f;

#pragma unroll
        for (int dy = 0; dy < 3; ++dy) {
            const float* row = &s_in[(r0 + dy) * WR * CG + ch];
            float a = row[0];
            float bb = row[CG];
#pragma unroll
            for (int pc = 0; pc < TW; ++pc) {
                float c = row[(pc + 2) * CG];
                accs[pc] = fmaf(a,  k[dy * 3 + 0], accs[pc]);
                accs[pc] = fmaf(bb, k[dy * 3 + 1], accs[pc]);
                accs[pc] = fmaf(c,  k[dy * 3 + 2], accs[pc]);
                a = bb; bb = c;
            }
        }
#pragma unroll
        for (int pc = 0; pc < TW; ++pc)
            s_dw[(r0 * TW + pc) * CG + ch] = accs[pc];
    }

    __syncthreads();

    // ---------------- pointwise via V_WMMA_F32_16X16X4_F32 -----------------------
    // Per wave: M = 16 pixels, K = 32 (8 chained k=4 WMMAs), N = 32 (2 halves).
    const int wave  = tid >> 5;
    const int lane  = tid & 31;
    const int mrow  = lane & 15;            // M (A) / N (B,C,D) within 16
    const int kup   = (lane >> 4) * 2;      // upper half-wave holds K+2,K+3
    const int mbase = wave * 16;            // 8 waves x 16 pixels = 128-pixel tile

    v8f acc[2];
#pragma unroll
    for (int nh = 0; nh < 2; ++nh) {
        const int ncol = nh * 16 + mrow;    // output channel within group
        v8f a8 = {0.f, 0.f, 0.f, 0.f, 0.f, 0.f, 0.f, 0.f};
#pragma unroll
        for (int ks = 0; ks < 8; ++ks) {
            const int k0 = ks * 4 + kup;
            // A fragment: lane holds (M=mrow, K=k0..k0+1) -> one ds_load_b64
            v2f a = *(const v2f*)&s_dw[(mbase + mrow) * CG + k0];
            // B fragment: lane holds (K=k0..k0+1, N=ncol) -> one ds_load_b64
            v2f bm = *(const v2f*)&s_pw[((k0 >> 1) * CG + ncol) * 2];
            a8 = __builtin_amdgcn_wmma_f32_16x16x4_f32(
                false, a, false, bm, (short)0, a8, false, false);
        }
        acc[nh] = a8;
    }

    // ---------------- BN + LeakyReLU, restage into LDS (pixel-major) -------------
    // Both accumulators are complete, so overwriting s_dw is safe: each wave
    // reads and writes only its own 16 rows of s_dw.
    const int rowOff = (lane >> 4) * 8;
#pragma unroll
    for (int nh = 0; nh < 2; ++nh) {
        const int ncol = nh * 16 + mrow;
        const int cg   = chBase + ncol;
        const float sc = gamma_[cg] * RSQ(mvar[cg] + BN_EPS);
        const float bi = fmaf(-mmean[cg], sc, beta_[cg]);
#pragma unroll
        for (int r = 0; r < 8; ++r) {
            float v = fmaf(acc[nh][r], sc, bi);
            v = (v >= 0.f) ? v : LRELU_ALPHA * v;
            s_dw[(mbase + rowOff + r) * CG + ncol] = v;
        }
    }

    __syncthreads();

    // ---------------- write-out: 128B-contiguous per pixel ----------------------
    // LDS -> global async b128 (no VGPR round-trip); S_ENDPGM's implicit
    // wait-idle covers ASYNCcnt completion.
    for (int idx = tid; idx < TH * TW * (CG / 4); idx += 256) {
        int p = idx >> 3;               // pixel in tile
        int q = idx & 7;                // float4-quad within group channels
        int hh = th0 + (p >> 4);
        int ww = tw0 + (p & 15);
        float* lp = &s_dw[p * CG + q * 4];
        float* gp = out + (((size_t)b * HH + hh) * WW + ww) * CC + chBase + q * 4;
#if USE_ASYNC_ST
        __builtin_amdgcn_global_store_async_from_lds_b128(
            (glb_v4i_ptr)gp, (lds_v4i_ptr)lp, 0, 0);
#else
        v4f v = *(const v4f*)lp;
        __builtin_nontemporal_store(v, (v4f*)gp);
#endif
    }
}

extern "C" void kernel_launch(void* const* d_in, const int* in_sizes, int n_in,
                              void* d_out, int out_size, void* d_ws, size_t ws_size,
                              hipStream_t stream) {
    (void)in_sizes; (void)n_in; (void)out_size; (void)d_ws; (void)ws_size;
    const float* x     = (const float*)d_in[0];
    const float* dwk   = (const float*)d_in[1];
    const float* pwk   = (const float*)d_in[2];
    const float* gamma = (const float*)d_in[3];
    const float* beta  = (const float*)d_in[4];
    const float* mmean = (const float*)d_in[5];
    const float* mvar  = (const float*)d_in[6];
    float* out = (float*)d_out;

    dim3 grid((HH / TH) * (WW / TW), BB, GG);   // 32 x 16 x 32 blocks
    fused_dw_pw_bn_lrelu<<<grid, 256, 0, stream>>>(x, dwk, pwk, gamma, beta,
                                                   mmean, mvar, out);
}